// Augmentation_59176059404649
// MI455X (gfx1250) — compile-verified
//
#include <hip/hip_runtime.h>
#include <hip/hip_bf16.h>
#include <stdint.h>

// Problem constants from the reference
#define B_   32
#define N_   8192
#define DIM_ 256
#define M_   1228
#define ROWS_      (B_ * N_)       // 262144 rows of 256 floats (1 KB each)
#define ROW_F4_    (DIM_ / 4)      // 64 float4 per row
#define FLAG_WORDS (ROWS_ / 32)    // 8192 uint32 words (32 KB bitmap)

typedef float f4  __attribute__((ext_vector_type(4)));
typedef int   v4i __attribute__((ext_vector_type(4)));

#if defined(__HIP_DEVICE_COMPILE__)
typedef __attribute__((address_space(1))) v4i* global_v4i_ptr;
typedef __attribute__((address_space(3))) v4i* lds_v4i_ptr;
#endif

// ---------------------------------------------------------------------------
// Kernel 1: clear the row-mask bitmap (d_ws is not re-poisoned between calls,
// so we must re-init every launch for determinism).
// ---------------------------------------------------------------------------
__global__ void clear_flags_kernel(uint32_t* __restrict__ flags) {
    int i = blockIdx.x * blockDim.x + threadIdx.x;
    if (i < FLAG_WORDS) flags[i] = 0u;
}

// ---------------------------------------------------------------------------
// Kernel 2: set one bit per masked row; also emit the second tuple output
// (mask_indices bit-copied as int32 into the tail of d_out).
// ---------------------------------------------------------------------------
__global__ void scatter_flags_kernel(const int* __restrict__ mask_indices,
                                     uint32_t* __restrict__ flags,
                                     int* __restrict__ out_idx_tail) {
    int i = blockIdx.x * blockDim.x + threadIdx.x;
    if (i < B_ * M_) {
        int idx = mask_indices[i];          // column index within row, [0, N)
        out_idx_tail[i] = idx;              // tuple output #2 (identity copy)
        int b   = i / M_;
        int row = b * N_ + idx;
        atomicOr(&flags[row >> 5], 1u << (row & 31));
    }
}

// ---------------------------------------------------------------------------
// Kernel 3: single-pass masked copy. One wave32 per 1 KB row: lane l moves
// float4 l and l+32. The masked/unmasked decision is wave-uniform -> no
// divergence. Masked rows never touch x in HBM; they stream emb from LDS.
// NT hints keep the 512 MB stream from thrashing the 192 MB L2.
// ---------------------------------------------------------------------------
__global__ void mask_copy_kernel(const f4* __restrict__ x,
                                 const f4* __restrict__ emb,
                                 const uint32_t* __restrict__ flags,
                                 f4* __restrict__ out) {
    __shared__ f4 semb[ROW_F4_];   // 1 KB: the broadcast emb_mask row

    const int lane = threadIdx.x & 31;
    const int wave = threadIdx.x >> 5;     // 0..7 (256 threads = 8 waves)

    // Stage emb_mask into LDS. Preferred path: CDNA5 async global->LDS DMA
    // (tracked by ASYNCcnt), issued by wave 0 with full EXEC.
#if defined(__HIP_DEVICE_COMPILE__) && defined(__has_builtin) && \
    __has_builtin(__builtin_amdgcn_global_load_async_to_lds_b128)
    if (threadIdx.x < 32) {
        __builtin_amdgcn_global_load_async_to_lds_b128(
            (global_v4i_ptr)(emb + lane),
            (lds_v4i_ptr)(semb + lane),
            0, 0);
        __builtin_amdgcn_global_load_async_to_lds_b128(
            (global_v4i_ptr)(emb + lane + 32),
            (lds_v4i_ptr)(semb + lane + 32),
            0, 0);
#if __has_builtin(__builtin_amdgcn_s_wait_asynccnt)
        __builtin_amdgcn_s_wait_asynccnt(0);
#else
        asm volatile("s_wait_asynccnt 0" ::: "memory");
#endif
    }
    __syncthreads();
#else
    if (threadIdx.x < ROW_F4_) semb[threadIdx.x] = emb[threadIdx.x];
    __syncthreads();
#endif

    const int groups = ROWS_ / 8;          // 8 rows (one per wave) per block-iter
    for (int g = blockIdx.x; g < groups; g += gridDim.x) {
        const int row = g * 8 + wave;      // wave-uniform row id
        const uint32_t w = flags[row >> 5];
        const bool masked = (w >> (row & 31)) & 1u;

        f4* __restrict__ dst = out + (size_t)row * ROW_F4_;
        if (masked) {
            // Broadcast row from LDS; skip the HBM read of x entirely.
            __builtin_nontemporal_store(semb[lane],      dst + lane);
            __builtin_nontemporal_store(semb[lane + 32], dst + lane + 32);
        } else {
            const f4* __restrict__ src = x + (size_t)row * ROW_F4_;
            f4 a = __builtin_nontemporal_load(src + lane);
            f4 b = __builtin_nontemporal_load(src + lane + 32);
            __builtin_nontemporal_store(a, dst + lane);
            __builtin_nontemporal_store(b, dst + lane + 32);
        }
    }
}

// ---------------------------------------------------------------------------
// Launcher
// ---------------------------------------------------------------------------
extern "C" void kernel_launch(void* const* d_in, const int* in_sizes, int n_in,
                              void* d_out, int out_size, void* d_ws, size_t ws_size,
                              hipStream_t stream) {
    const f4*  x    = (const f4*)d_in[0];          // [B, N, DIM] f32
    const int* midx = (const int*)d_in[1];         // [B, M] int32
    const f4*  emb  = (const f4*)d_in[2];          // [1, DIM] f32

    f4*       out_x    = (f4*)d_out;                           // tuple output #1
    int*      out_tail = (int*)d_out + (size_t)ROWS_ * DIM_;   // tuple output #2
    uint32_t* flags    = (uint32_t*)d_ws;                      // 32 KB bitmap

    // 1) clear bitmap (8192 words)
    clear_flags_kernel<<<(FLAG_WORDS + 255) / 256, 256, 0, stream>>>(flags);

    // 2) set bits for masked rows + copy indices to output tail
    scatter_flags_kernel<<<(B_ * M_ + 255) / 256, 256, 0, stream>>>(
        midx, flags, out_tail);

    // 3) one-pass masked copy: 4096 persistent blocks x 8 waves, 8 iterations
    mask_copy_kernel<<<4096, 256, 0, stream>>>(x, emb, flags, out_x);
}